// HAN_77335181132166
// MI455X (gfx1250) — compile-verified
//
#include <hip/hip_runtime.h>
#include <math.h>

#define NC 100000
#define NP 200000
#define NE 1000000
#define DIN 64
#define CCH 128
#define NHD 8
#define NEG_SLOPE 0.2f
#define BN_EPS 1e-5f

typedef __attribute__((ext_vector_type(2))) float v2f;
typedef __attribute__((ext_vector_type(8))) float v8f;
typedef __attribute__((ext_vector_type(4))) int   v4i;

#if defined(__has_builtin)
#if __has_builtin(__builtin_amdgcn_global_load_async_to_lds_b128) && \
    __has_builtin(__builtin_amdgcn_s_wait_asynccnt)
#define HAS_ASYNC_LDS 1
#endif
#endif

#ifdef HAS_ASYNC_LDS
typedef __attribute__((address_space(1))) v4i* gptr_v4i;  // global (AS1)
typedef __attribute__((address_space(3))) v4i* lptr_v4i;  // LDS (AS3)
#endif

// ---------------- helpers ----------------

__device__ __forceinline__ float atomicMaxF(float* addr, float val) {
    // IEEE-ordered atomic max: int compare works for >=0, unsigned-min for
    // negatives (init value is -inf = 0xFF800000).
    if (val >= 0.0f)
        return __int_as_float(atomicMax((int*)addr, __float_as_int(val)));
    else
        return __uint_as_float(atomicMin((unsigned int*)addr, __float_as_uint(val)));
}

__global__ void fill_kernel(float* __restrict__ p, float v, size_t n) {
    size_t i = (size_t)blockIdx.x * blockDim.x + threadIdx.x;
    if (i < n) p[i] = v;
}

__global__ void relu_kernel(float* __restrict__ p, size_t n) {
    size_t i = (size_t)blockIdx.x * blockDim.x + threadIdx.x;
    if (i < n) p[i] = fmaxf(p[i], 0.0f);
}

// -------- WMMA fp32 GEMM, LDS-staged weights, 16xN strip per wave --------
// Block = 256 threads = 8 waves; each wave computes a 16-row x (NT*16)-col
// output strip. W[K x NT*16] is staged once per workgroup into LDS via the
// gfx1250 async global->LDS path (fallback: float4 copies -> ds_store).
// A fragment (16x4 fp32, 2 VGPRs) is loaded once per k-step and reused for
// all NT WMMAs. mode 0: Cout = A@W + bias. mode 1: atomicAdd column-sums of
// tanh(A@W + bias) into ksum (semantic attention, no Cout written).

template <int NT>
__global__ void gemm_wmma_tiled(const float* __restrict__ A,
                                const float* __restrict__ W,
                                const float* __restrict__ bias,
                                float* __restrict__ Cout,
                                float* __restrict__ ksum,
                                int M, int K, int mode) {
    extern __shared__ float ldsW[];
    const int N = NT * 16;
    const int tid = threadIdx.x;

    // ---- stage W[K*N] into LDS ----
    const int nchunk = (K * N) >> 2;  // float4 chunks
#ifdef HAS_ASYNC_LDS
    for (int c = tid; c < nchunk; c += 256) {
        __builtin_amdgcn_global_load_async_to_lds_b128(
            (gptr_v4i)(W + 4 * c), (lptr_v4i)(ldsW + 4 * c), 0, 0);
    }
    __builtin_amdgcn_s_wait_asynccnt(0);
#else
    for (int c = tid; c < nchunk; c += 256) {
        ((float4*)ldsW)[c] = ((const float4*)W)[c];
    }
#endif
    __syncthreads();

    const int wave = tid >> 5;
    const int lane = tid & 31;
    const int half = lane >> 4;
    const int l15  = lane & 15;
    const int tm = (blockIdx.x * 8 + wave) * 16;
    if (tm >= M) return;  // uniform per-wave guard (after barrier)

    v8f acc[NT];
#pragma unroll
    for (int t = 0; t < NT; ++t) acc[t] = {};

    const float* Arow = A + (size_t)(tm + l15) * K;
    for (int k = 0; k < K; k += 4) {
        v2f a;
        a.x = Arow[k + 2 * half];      // one b64 load: K={k,k+1} / {k+2,k+3}
        a.y = Arow[k + 2 * half + 1];
        const float* w0 = ldsW + (k + 2 * half) * N + l15;
#pragma unroll
        for (int t = 0; t < NT; ++t) {
            v2f b;
            b.x = w0[t * 16];          // ds_load: row k+2*half
            b.y = w0[t * 16 + N];      // ds_load: row k+2*half+1
            acc[t] = __builtin_amdgcn_wmma_f32_16x16x4_f32(
                         false, a, false, b, (short)0, acc[t], false, false);
        }
    }

    if (mode == 0) {
#pragma unroll
        for (int t = 0; t < NT; ++t) {
            int col = t * 16 + l15;
            float bcol = bias[col];
#pragma unroll
            for (int j = 0; j < 8; ++j) {
                int row = tm + half * 8 + j;
                Cout[(size_t)row * N + col] = acc[t][j] + bcol;
            }
        }
    } else {
#pragma unroll
        for (int t = 0; t < NT; ++t) {
            int col = t * 16 + l15;
            float bcol = bias[col];
            float partial = 0.0f;
#pragma unroll
            for (int j = 0; j < 8; ++j) partial += tanhf(acc[t][j] + bcol);
            partial += __shfl_xor(partial, 16, 32);  // merge row halves
            if (half == 0) atomicAdd(&ksum[col], partial);
        }
    }
}

// ---------------- attention phase ----------------

__global__ void node_logits_kernel(const float* __restrict__ Hn,
                                   const float* __restrict__ avec,
                                   float* __restrict__ al, int N) {
    int t = blockIdx.x * blockDim.x + threadIdx.x;
    if (t >= N * NHD) return;
    int n = t >> 3, h = t & 7;
    const float* hp = Hn + (size_t)n * CCH + h * 16;
    const float* ap = avec + h * 16;
    float s = 0.0f;
#pragma unroll
    for (int d = 0; d < 16; ++d) s += hp[d] * ap[d];
    al[t] = s;
}

__global__ void edge_logits_max_kernel(const float* __restrict__ al_src,
                                       const float* __restrict__ al_dst,
                                       const int* __restrict__ src,
                                       const int* __restrict__ dst,
                                       float* __restrict__ eal,
                                       float* __restrict__ m, int E) {
    int t = blockIdx.x * blockDim.x + threadIdx.x;
    if (t >= E * NHD) return;
    int e = t >> 3, h = t & 7;
    float v = al_src[src[e] * NHD + h] + al_dst[dst[e] * NHD + h];
    v = (v >= 0.0f) ? v : NEG_SLOPE * v;
    eal[t] = v;
    atomicMaxF(&m[dst[e] * NHD + h], v);
}

__global__ void edge_exp_sum_kernel(float* __restrict__ eal,
                                    const int* __restrict__ dst,
                                    const float* __restrict__ m,
                                    float* __restrict__ ssum, int E) {
    int t = blockIdx.x * blockDim.x + threadIdx.x;
    if (t >= E * NHD) return;
    int e = t >> 3, h = t & 7;
    float ex = __expf(eal[t] - m[dst[e] * NHD + h]);
    eal[t] = ex;
    atomicAdd(&ssum[dst[e] * NHD + h], ex);
}

__global__ void edge_msg_kernel(const float* __restrict__ Hsrc,
                                const float* __restrict__ eal,
                                const float* __restrict__ ssum,
                                const int* __restrict__ src,
                                const int* __restrict__ dst,
                                float* __restrict__ outp, int E) {
    int t = blockIdx.x * blockDim.x + threadIdx.x;
    if (t >= E * 32) return;
    int e = t >> 5, lane = t & 31;
    int c0 = lane * 4, h = c0 >> 4;
    int si = src[e], di = dst[e];
    float alpha = eal[(size_t)e * NHD + h] /
                  (ssum[(size_t)di * NHD + h] + 1e-16f);
    const float* hs = Hsrc + (size_t)si * CCH + c0;
    float* op = outp + (size_t)di * CCH + c0;
#pragma unroll
    for (int i = 0; i < 4; ++i) atomicAdd(&op[i], hs[i] * alpha);
}

// ---------------- semantic attention + MLP head ----------------

__global__ void sem_attn_kernel(const float* __restrict__ ksum,
                                const float* __restrict__ q,
                                float* __restrict__ attn) {
    if (threadIdx.x == 0 && blockIdx.x == 0) {
        float s0 = 0.0f, s1 = 0.0f;
        for (int c = 0; c < CCH; ++c) {
            s0 += (ksum[c] / (float)NP) * q[c];
            s1 += (ksum[CCH + c] / (float)NP) * q[c];
        }
        float mx = fmaxf(s0, s1);
        float e0 = __expf(s0 - mx), e1 = __expf(s1 - mx);
        float inv = 1.0f / (e0 + e1);
        attn[0] = e0 * inv;
        attn[1] = e1 * inv;
    }
}

__global__ void combine_kernel(const float* __restrict__ a,
                               const float* __restrict__ b,
                               const float* __restrict__ attn,
                               float* __restrict__ out, size_t n) {
    size_t i = (size_t)blockIdx.x * blockDim.x + threadIdx.x;
    if (i < n) out[i] = attn[0] * a[i] + attn[1] * b[i];
}

__global__ void colstats_kernel(const float* __restrict__ X,
                                float* __restrict__ sum,
                                float* __restrict__ sumsq, int M, int N) {
    int t = blockIdx.x * blockDim.x + threadIdx.x;
    int c = t % N;
    int stride = (gridDim.x * blockDim.x) / N;
    float s = 0.0f, s2 = 0.0f;
    for (int r = t / N; r < M; r += stride) {
        float v = X[(size_t)r * N + c];
        s += v;
        s2 += v * v;
    }
    atomicAdd(&sum[c], s);
    atomicAdd(&sumsq[c], s2);
}

__global__ void bn_relu_kernel(const float* __restrict__ X,
                               const float* __restrict__ sum,
                               const float* __restrict__ sumsq,
                               const float* __restrict__ g,
                               const float* __restrict__ be,
                               float* __restrict__ Y, int M, int N) {
    size_t i = (size_t)blockIdx.x * blockDim.x + threadIdx.x;
    if (i >= (size_t)M * N) return;
    int c = (int)(i % N);
    float mu = sum[c] / (float)M;
    float var = sumsq[c] / (float)M - mu * mu;
    float v = g[c] * (X[i] - mu) * rsqrtf(var + BN_EPS) + be[c];
    Y[i] = fmaxf(v, 0.0f);
}

__global__ void final_head_kernel(const float* __restrict__ H2,
                                  const float* __restrict__ W3,
                                  const float* __restrict__ b3,
                                  float* __restrict__ out, int M) {
    int r = blockIdx.x * blockDim.x + threadIdx.x;
    if (r >= M) return;
    const float* hp = H2 + (size_t)r * 64;
    float s = b3[0];
#pragma unroll
    for (int k = 0; k < 64; ++k) s += hp[k] * W3[k];
    out[r] = 1.0f / (1.0f + __expf(-s));
}

// ---------------- orchestration ----------------

extern "C" void kernel_launch(void* const* d_in, const int* in_sizes, int n_in,
                              void* d_out, int out_size, void* d_ws, size_t ws_size,
                              hipStream_t stream) {
    const float* x_c      = (const float*)d_in[0];
    const float* x_p      = (const float*)d_in[1];
    const int*   e_src_cp = (const int*)d_in[2];
    const int*   e_dst_cp = (const int*)d_in[3];
    const int*   e_src_pp = (const int*)d_in[4];
    const int*   e_dst_pp = (const int*)d_in[5];
    const float* W_c = (const float*)d_in[6];
    const float* b_c = (const float*)d_in[7];
    const float* W_p = (const float*)d_in[8];
    const float* b_p = (const float*)d_in[9];
    const float* a_src_cp = (const float*)d_in[10];
    const float* a_dst_cp = (const float*)d_in[11];
    const float* a_src_pp = (const float*)d_in[12];
    const float* a_dst_pp = (const float*)d_in[13];
    const float* Wk    = (const float*)d_in[14];
    const float* bk    = (const float*)d_in[15];
    const float* q_sem = (const float*)d_in[16];
    const float* W1 = (const float*)d_in[17];
    const float* b1 = (const float*)d_in[18];
    const float* g1 = (const float*)d_in[19];
    const float* be1 = (const float*)d_in[20];
    const float* W2 = (const float*)d_in[21];
    const float* b2 = (const float*)d_in[22];
    const float* g2 = (const float*)d_in[23];
    const float* be2 = (const float*)d_in[24];
    const float* W3 = (const float*)d_in[25];
    const float* b3 = (const float*)d_in[26];
    float* out = (float*)d_out;
    float* ws = (float*)d_ws;

    // workspace layout (floats); big buffers reused across phases
    float* h_c      = ws;                                   // 12.8M
    float* h_p      = h_c + (size_t)NC * CCH;               // 25.6M
    float* out_cp   = h_p + (size_t)NP * CCH;               // 25.6M
    float* out_pp   = out_cp + (size_t)NP * CCH;            // 25.6M
    float* eal_cp   = out_pp + (size_t)NP * CCH;            // 8M
    float* eal_pp   = eal_cp + (size_t)NE * NHD;            // 8M
    float* al_src_c = eal_pp + (size_t)NE * NHD;            // 0.8M
    float* al_dst_c = al_src_c + (size_t)NC * NHD;          // 1.6M
    float* al_src_p = al_dst_c + (size_t)NP * NHD;          // 1.6M
    float* al_dst_p = al_src_p + (size_t)NP * NHD;          // 1.6M
    float* m_cp = al_dst_p + (size_t)NP * NHD;              // 1.6M
    float* s_cp = m_cp + (size_t)NP * NHD;
    float* m_pp = s_cp + (size_t)NP * NHD;
    float* s_pp = m_pp + (size_t)NP * NHD;
    float* ksum  = s_pp + (size_t)NP * NHD;                 // 256
    float* attn  = ksum + 256;                              // 2
    float* stats = attn + 2;                                // 384
    // reuse after edge phase / combine:
    float* comb = h_p;     // semantic-combined features
    float* z1   = out_cp;  // MLP layer-1 pre-BN
    float* h1   = out_pp;  // MLP layer-1 post-BN
    float* z2   = eal_cp;  // MLP layer-2 pre-BN (12.8M fits in 16M eal region)
    float* h2   = h_c;     // MLP layer-2 post-BN

    const int T = 256;
    const size_t npc = (size_t)NP * CCH;
    const int gNC = (NC / 16 + 7) / 8;   // 782 row-strip blocks
    const int gNP = (NP / 16 + 7) / 8;   // 1563 row-strip blocks

    // ---- node projections (WMMA fp32, async-LDS staged weights) ----
    gemm_wmma_tiled<8><<<gNC, 256, DIN * CCH * sizeof(float), stream>>>(
        x_c, W_c, b_c, h_c, nullptr, NC, DIN, 0);
    gemm_wmma_tiled<8><<<gNP, 256, DIN * CCH * sizeof(float), stream>>>(
        x_p, W_p, b_p, h_p, nullptr, NP, DIN, 0);

    // ---- per-node attention logits ----
    node_logits_kernel<<<(NC * NHD + T - 1) / T, T, 0, stream>>>(h_c, a_src_cp, al_src_c, NC);
    node_logits_kernel<<<(NP * NHD + T - 1) / T, T, 0, stream>>>(h_p, a_dst_cp, al_dst_c, NP);
    node_logits_kernel<<<(NP * NHD + T - 1) / T, T, 0, stream>>>(h_p, a_src_pp, al_src_p, NP);
    node_logits_kernel<<<(NP * NHD + T - 1) / T, T, 0, stream>>>(h_p, a_dst_pp, al_dst_p, NP);

    // ---- init accumulators ----
    fill_kernel<<<((size_t)NP * NHD + T - 1) / T, T, 0, stream>>>(m_cp, -INFINITY, (size_t)NP * NHD);
    fill_kernel<<<((size_t)NP * NHD + T - 1) / T, T, 0, stream>>>(m_pp, -INFINITY, (size_t)NP * NHD);
    fill_kernel<<<((size_t)NP * NHD + T - 1) / T, T, 0, stream>>>(s_cp, 0.0f, (size_t)NP * NHD);
    fill_kernel<<<((size_t)NP * NHD + T - 1) / T, T, 0, stream>>>(s_pp, 0.0f, (size_t)NP * NHD);
    fill_kernel<<<(npc + T - 1) / T, T, 0, stream>>>(out_cp, 0.0f, npc);
    fill_kernel<<<(npc + T - 1) / T, T, 0, stream>>>(out_pp, 0.0f, npc);
    fill_kernel<<<(256 + 2 + 384 + T - 1) / T, T, 0, stream>>>(ksum, 0.0f, 256 + 2 + 384);

    // ---- edge softmax attention: customer->product ----
    edge_logits_max_kernel<<<(NE * NHD + T - 1) / T, T, 0, stream>>>(
        al_src_c, al_dst_c, e_src_cp, e_dst_cp, eal_cp, m_cp, NE);
    edge_exp_sum_kernel<<<(NE * NHD + T - 1) / T, T, 0, stream>>>(
        eal_cp, e_dst_cp, m_cp, s_cp, NE);
    edge_msg_kernel<<<(NE * 32 + T - 1) / T, T, 0, stream>>>(
        h_c, eal_cp, s_cp, e_src_cp, e_dst_cp, out_cp, NE);

    // ---- edge softmax attention: product->product ----
    edge_logits_max_kernel<<<(NE * NHD + T - 1) / T, T, 0, stream>>>(
        al_src_p, al_dst_p, e_src_pp, e_dst_pp, eal_pp, m_pp, NE);
    edge_exp_sum_kernel<<<(NE * NHD + T - 1) / T, T, 0, stream>>>(
        eal_pp, e_dst_pp, m_pp, s_pp, NE);
    edge_msg_kernel<<<(NE * 32 + T - 1) / T, T, 0, stream>>>(
        h_p, eal_pp, s_pp, e_src_pp, e_dst_pp, out_pp, NE);

    relu_kernel<<<(npc + T - 1) / T, T, 0, stream>>>(out_cp, npc);
    relu_kernel<<<(npc + T - 1) / T, T, 0, stream>>>(out_pp, npc);

    // ---- semantic attention (WMMA GEMM fused with tanh + column mean) ----
    gemm_wmma_tiled<8><<<gNP, 256, CCH * CCH * sizeof(float), stream>>>(
        out_cp, Wk, bk, nullptr, ksum, NP, CCH, 1);
    gemm_wmma_tiled<8><<<gNP, 256, CCH * CCH * sizeof(float), stream>>>(
        out_pp, Wk, bk, nullptr, ksum + CCH, NP, CCH, 1);
    sem_attn_kernel<<<1, 32, 0, stream>>>(ksum, q_sem, attn);
    combine_kernel<<<(npc + T - 1) / T, T, 0, stream>>>(out_cp, out_pp, attn, comb, npc);

    // ---- MLP head: layer 1 (128->128) + BN + ReLU ----
    gemm_wmma_tiled<8><<<gNP, 256, CCH * CCH * sizeof(float), stream>>>(
        comb, W1, b1, z1, nullptr, NP, CCH, 0);
    colstats_kernel<<<128, T, 0, stream>>>(z1, stats, stats + 128, NP, 128);
    bn_relu_kernel<<<(npc + T - 1) / T, T, 0, stream>>>(
        z1, stats, stats + 128, g1, be1, h1, NP, 128);

    // ---- layer 2 (128->64) + BN + ReLU ----
    gemm_wmma_tiled<4><<<gNP, 256, CCH * 64 * sizeof(float), stream>>>(
        h1, W2, b2, z2, nullptr, NP, CCH, 0);
    colstats_kernel<<<64, T, 0, stream>>>(z2, stats + 256, stats + 320, NP, 64);
    bn_relu_kernel<<<((size_t)NP * 64 + T - 1) / T, T, 0, stream>>>(
        z2, stats + 256, stats + 320, g2, be2, h2, NP, 64);

    // ---- final 64->1 + sigmoid ----
    final_head_kernel<<<(NP + T - 1) / T, T, 0, stream>>>(h2, W3, b3, out, NP);
}